// SO3reparameterize_21930103013450
// MI455X (gfx1250) — compile-verified
//
#include <hip/hip_runtime.h>
#include <hip/hip_bf16.h>

typedef __attribute__((ext_vector_type(2))) float v2f;
typedef __attribute__((ext_vector_type(4))) float v4f;
typedef __attribute__((ext_vector_type(8))) float v8f;

// ---------------------------------------------------------------------------
// Pack W_mu/W_d/W_l (each 3x1024) into WMMA B-fragment order for
// V_WMMA_F32_16X16X4_F32, padded to 16 columns with zeros.
// B-matrix (4x16, K x N) per-lane layout (wave32):
//   lane n      (n<16): VGPR0 = B[K=0][n], VGPR1 = B[K=1][n]
//   lane n+16        : VGPR0 = B[K=2][n], VGPR1 = B[K=3][n]
// Stored as Bpack[kk*64 + lane*2 + slot], kk = k/4.  Total 16384 floats (64KB).
// ---------------------------------------------------------------------------
__global__ __launch_bounds__(256) void pack_weights_kernel(
    const float* __restrict__ Wmu, const float* __restrict__ Wd,
    const float* __restrict__ Wl, float* __restrict__ Bpack) {
  int idx = blockIdx.x * 256 + threadIdx.x;   // 0 .. 16383
  int k = idx >> 4;
  int n = idx & 15;
  float val = 0.0f;
  if (n < 3)      val = Wmu[n * 1024 + k];
  else if (n < 6) val = Wd[(n - 3) * 1024 + k];
  else if (n < 9) val = Wl[(n - 6) * 1024 + k];
  int kk   = k >> 2;
  int r    = k & 3;
  int lane = n + ((r >= 2) ? 16 : 0);
  Bpack[kk * 64 + lane * 2 + (r & 1)] = val;
}

// ---------------------------------------------------------------------------
// Rodrigues exponential map, matching the reference exactly (no theta guard):
//   k = v/|v|, R = I + sin(th)*hat(k) + (1-cos(th))*(k k^T - I)
// ---------------------------------------------------------------------------
__device__ __forceinline__ void rodrigues(float vx, float vy, float vz,
                                          float R[9]) {
  float th  = sqrtf(vx * vx + vy * vy + vz * vz);
  float inv = 1.0f / th;
  float kx = vx * inv, ky = vy * inv, kz = vz * inv;
  float s  = sinf(th);
  float c1 = 1.0f - cosf(th);
  R[0] = 1.0f + c1 * (kx * kx - 1.0f);
  R[1] = -s * kz + c1 * kx * ky;
  R[2] =  s * ky + c1 * kx * kz;
  R[3] =  s * kz + c1 * kx * ky;
  R[4] = 1.0f + c1 * (ky * ky - 1.0f);
  R[5] = -s * kx + c1 * ky * kz;
  R[6] = -s * ky + c1 * kx * kz;
  R[7] =  s * kx + c1 * ky * kz;
  R[8] = 1.0f + c1 * (kz * kz - 1.0f);
}

__device__ __forceinline__ float softplus(float t) {
  return fmaxf(t, 0.0f) + log1pf(expf(-fabsf(t)));
}

// ---------------------------------------------------------------------------
// Fused GEMM (fp32 WMMA 16x16x4) + SO(3) reparameterize epilogue.
// 256 threads = 8 waves; wave w handles 16 rows; block covers 128 rows.
// ---------------------------------------------------------------------------
__global__ __launch_bounds__(256) void so3_fused_kernel(
    const float* __restrict__ x, const float* __restrict__ eps,
    const float* __restrict__ Bpack,
    const float* __restrict__ bmu, const float* __restrict__ bd,
    const float* __restrict__ bl,
    float* __restrict__ out, int Brows) {
  __shared__ float sB[16384];  // 64KB: B fragments, later reused as tile buffer

  const int tid = threadIdx.x;

  // Stage packed B into LDS with coalesced 128-bit loads (16 per thread).
  {
    const v4f* g  = (const v4f*)Bpack;
    v4f*       sh = (v4f*)sB;
#pragma unroll
    for (int i = 0; i < 16; ++i) sh[tid + 256 * i] = g[tid + 256 * i];
  }
  __syncthreads();

  const int wave = tid >> 5;
  const int lane = tid & 31;
  const int half = lane >> 4;   // 0: K%4 in {0,1}; 1: K%4 in {2,3}
  const int lrow = lane & 15;
  const long long row = (long long)blockIdx.x * 128 + wave * 16 + lrow;

  // A-matrix (16x4 f32) per-lane layout: lane l<16 holds x[row][k],x[row][k+1];
  // lane l+16 holds x[row][k+2],x[row][k+3].
  const float* xrow  = x + row * 1024 + half * 2;
  const float* bfrag = sB + lane * 2;

  v8f acc0 = {};
  v8f acc1 = {};
  for (int kk0 = 0; kk0 < 256; kk0 += 8) {
    // Stream-ahead prefetch (~2KB ahead); speculative, OOB is dropped.
    __builtin_prefetch(xrow + kk0 * 4 + 512, 0, 3);
#pragma unroll
    for (int u = 0; u < 8; u += 2) {
      const int kk = kk0 + u;
      v2f a0 = *(const v2f*)(xrow + kk * 4);
      v2f b0 = *(const v2f*)(bfrag + kk * 64);
      v2f a1 = *(const v2f*)(xrow + kk * 4 + 4);
      v2f b1 = *(const v2f*)(bfrag + kk * 64 + 64);
      acc0 = __builtin_amdgcn_wmma_f32_16x16x4_f32(
          false, a0, false, b0, (short)0, acc0, false, false);
      acc1 = __builtin_amdgcn_wmma_f32_16x16x4_f32(
          false, a1, false, b1, (short)0, acc1, false, false);
    }
  }
  v8f acc = acc0 + acc1;

  // All waves are done reading sB; reuse it as the per-wave transpose tile.
  __syncthreads();
  float* tw = sB + wave * 288;  // 16 rows x stride 18 (bank-conflict-free)
#pragma unroll
  for (int j = 0; j < 8; ++j) {
    // C/D layout: VGPR j, lanes 0-15 -> M=j, lanes 16-31 -> M=j+8; N=lane&15.
    tw[(half * 8 + j) * 18 + lrow] = acc[j];
  }
  __syncthreads();

  // ---- Epilogue: each lane handles row `lrow`, samples {2*half, 2*half+1} --
  float h[9];
#pragma unroll
  for (int j = 0; j < 9; ++j) h[j] = tw[lrow * 18 + j];

  const float mu0 = h[0] + bmu[0];
  const float mu1 = h[1] + bmu[1];
  const float mu2 = h[2] + bmu[2];
  const float sd0 = sqrtf(softplus(h[3] + bd[0]));
  const float sd1 = sqrtf(softplus(h[4] + bd[1]));
  const float sd2 = sqrtf(softplus(h[5] + bd[2]));
  const float l0  = h[6] + bl[0];
  const float l1  = h[7] + bl[1];
  const float l2  = h[8] + bl[2];

  float Rm[9];
  rodrigues(mu0, mu1, mu2, Rm);

#pragma unroll
  for (int ss = 0; ss < 2; ++ss) {
    const int s = half * 2 + ss;
    const float* e = eps + ((long long)s * Brows + row) * 3;
    const float u0 = sd0 * e[0];
    const float u1 = sd1 * e[1];
    const float u2 = sd2 * e[2];
    // v = L @ u, L = [[1,0,0],[l0,1,0],[l1,l2,1]]
    const float w0 = u0;
    const float w1 = fmaf(l0, u0, u1);
    const float w2 = fmaf(l1, u0, fmaf(l2, u1, u2));
    float Rv[9];
    rodrigues(w0, w1, w2, Rv);
    float* o = out + ((long long)s * Brows + row) * 9;
#pragma unroll
    for (int i = 0; i < 3; ++i) {
#pragma unroll
      for (int j = 0; j < 3; ++j) {
        o[i * 3 + j] = fmaf(Rm[i * 3 + 0], Rv[0 + j],
                       fmaf(Rm[i * 3 + 1], Rv[3 + j],
                            Rm[i * 3 + 2] * Rv[6 + j]));
      }
    }
  }
}

extern "C" void kernel_launch(void* const* d_in, const int* in_sizes, int n_in,
                              void* d_out, int out_size, void* d_ws,
                              size_t ws_size, hipStream_t stream) {
  const float* x   = (const float*)d_in[0];
  const float* eps = (const float*)d_in[1];
  const float* Wmu = (const float*)d_in[2];
  const float* bmu = (const float*)d_in[3];
  const float* Wd  = (const float*)d_in[4];
  const float* bd  = (const float*)d_in[5];
  const float* Wl  = (const float*)d_in[6];
  const float* bl  = (const float*)d_in[7];

  const int Brows = in_sizes[0] / 1024;   // 131072
  float* Bpack = (float*)d_ws;            // 64KB scratch

  pack_weights_kernel<<<64, 256, 0, stream>>>(Wmu, Wd, Wl, Bpack);
  so3_fused_kernel<<<Brows / 128, 256, 0, stream>>>(
      x, eps, Bpack, bmu, bd, bl, (float*)d_out, Brows);
}